// CUDATopKRouter_31078383354368
// MI455X (gfx1250) — compile-verified
//
#include <hip/hip_runtime.h>
#include <stdint.h>

#define N_EXPERTS        64
#define TOP_K            8
#define WAVES_PER_BLOCK  4
#define BLOCK_THREADS    (WAVES_PER_BLOCK * 32)
#define TOKENS_PER_WAVE  32
#define TOKENS_PER_BLOCK (WAVES_PER_BLOCK * TOKENS_PER_WAVE)   // 128 tokens / block
#define ROW_STRIDE       68   // floats; 16B-aligned padded LDS row stride (bank-conflict ~2-way max)

// ---- CDNA5 async global->LDS path (gfx1250 only), with portable fallback ----
#if defined(__HIP_DEVICE_COMPILE__) && defined(__gfx1250__) && \
    __has_builtin(__builtin_amdgcn_global_load_async_to_lds_b128)
#define USE_ASYNC_LDS 1
#else
#define USE_ASYNC_LDS 0
#endif

#if USE_ASYNC_LDS
typedef int v4i __attribute__((ext_vector_type(4)));
typedef __attribute__((address_space(1))) v4i* g_v4i_ptr;   // global int4*
typedef __attribute__((address_space(3))) v4i* l_v4i_ptr;   // LDS int4*
#endif

// Compare-exchange: after call, (va,ia) holds the larger value (descending order).
// Strict '>' keeps original order on exact ties (lowest index first, like lax.top_k).
__device__ __forceinline__ void ce(float& va, int& ia, float& vb, int& ib) {
    const bool  c  = vb > va;
    const float tv = c ? vb : va;
    const int   ti = c ? ib : ia;
    vb = c ? va : vb;
    ib = c ? ia : ib;
    va = tv;
    ia = ti;
}

// Batcher odd-even mergesort for 8 elements, descending. 19 CEs.
__device__ __forceinline__ void sort8(float v[8], int id[8]) {
    ce(v[0],id[0],v[1],id[1]); ce(v[2],id[2],v[3],id[3]);
    ce(v[4],id[4],v[5],id[5]); ce(v[6],id[6],v[7],id[7]);
    ce(v[0],id[0],v[2],id[2]); ce(v[1],id[1],v[3],id[3]);
    ce(v[4],id[4],v[6],id[6]); ce(v[5],id[5],v[7],id[7]);
    ce(v[1],id[1],v[2],id[2]); ce(v[5],id[5],v[6],id[6]);
    ce(v[0],id[0],v[4],id[4]); ce(v[1],id[1],v[5],id[5]);
    ce(v[2],id[2],v[6],id[6]); ce(v[3],id[3],v[7],id[7]);
    ce(v[2],id[2],v[4],id[4]); ce(v[3],id[3],v[5],id[5]);
    ce(v[1],id[1],v[2],id[2]); ce(v[3],id[3],v[4],id[4]); ce(v[5],id[5],v[6],id[6]);
}

// Bitonic merge (clean) for a length-8 bitonic sequence -> descending. 12 CEs.
__device__ __forceinline__ void bitonic8(float v[8], int id[8]) {
    ce(v[0],id[0],v[4],id[4]); ce(v[1],id[1],v[5],id[5]);
    ce(v[2],id[2],v[6],id[6]); ce(v[3],id[3],v[7],id[7]);
    ce(v[0],id[0],v[2],id[2]); ce(v[1],id[1],v[3],id[3]);
    ce(v[4],id[4],v[6],id[6]); ce(v[5],id[5],v[7],id[7]);
    ce(v[0],id[0],v[1],id[1]); ce(v[2],id[2],v[3],id[3]);
    ce(v[4],id[4],v[5],id[5]); ce(v[6],id[6],v[7],id[7]);
}

extern __shared__ __align__(16) float smem[];   // TOKENS_PER_BLOCK * ROW_STRIDE floats

__global__ __launch_bounds__(BLOCK_THREADS)
void topk_router_kernel(const float* __restrict__ logits,
                        long long*   __restrict__ out_idx,
                        float*       __restrict__ out_w,
                        int n_tokens) {
    const int lane      = threadIdx.x & 31;
    const int wv        = threadIdx.x >> 5;
    const int tokenBase = blockIdx.x * TOKENS_PER_BLOCK + wv * TOKENS_PER_WAVE;

    float* wlds = smem + wv * (TOKENS_PER_WAVE * ROW_STRIDE);

    // ---- Stage this wave's 32 tokens (8 KB, contiguous in global) into LDS ----
    // Linear float f = t*128 + lane*4 maps to (tok = f/64, elem = f%64).
    const float* gbase       = logits + (size_t)tokenBase * N_EXPERTS;
    const int    tok_in_pair = lane >> 4;          // 0 or 1
    const int    elem4       = (lane & 15) * 4;    // 0..60

    if (tokenBase + TOKENS_PER_WAVE <= n_tokens) {
#if USE_ASYNC_LDS
        #pragma unroll
        for (int t = 0; t < 16; ++t) {
            const int tok = 2 * t + tok_in_pair;
            const float* g = gbase + t * 128 + lane * 4;          // coalesced 512B / instr
            float*       l = wlds + tok * ROW_STRIDE + elem4;     // 16B aligned (stride 68*4B)
            __builtin_amdgcn_global_load_async_to_lds_b128(
                (g_v4i_ptr)(uintptr_t)g,
                (l_v4i_ptr)(uint32_t)(uintptr_t)l,
                0, 0);
        }
#if __has_builtin(__builtin_amdgcn_s_wait_asynccnt)
        __builtin_amdgcn_s_wait_asynccnt(0);
#else
        asm volatile("s_wait_asynccnt 0" ::: "memory");
#endif
#else
        #pragma unroll
        for (int t = 0; t < 16; ++t) {
            const int tok = 2 * t + tok_in_pair;
            const float4 d = *(const float4*)(gbase + t * 128 + lane * 4);
            *(float4*)(wlds + tok * ROW_STRIDE + elem4) = d;
        }
#endif
    } else {
        // Tail (not hit for the reference size of 2^20 tokens)
        for (int t = 0; t < 16; ++t) {
            const int tok = 2 * t + tok_in_pair;
            if (tokenBase + tok < n_tokens) {
                const float4 d = *(const float4*)(gbase + t * 128 + lane * 4);
                *(float4*)(wlds + tok * ROW_STRIDE + elem4) = d;
            }
        }
    }
    // Each wave only reads the LDS region it wrote: no workgroup barrier needed.

    const int token = tokenBase + lane;
    if (token >= n_tokens) return;

    const float* row = wlds + lane * ROW_STRIDE;   // ds_load_b128, <=2-way bank conflicts

    // ---- Top-8 of 64 via sort-8 groups + bitonic top-k merges ----
    float rv[8]; int ri[8];
    {
        const float4 a = *(const float4*)(row + 0);
        const float4 b = *(const float4*)(row + 4);
        rv[0]=a.x; rv[1]=a.y; rv[2]=a.z; rv[3]=a.w;
        rv[4]=b.x; rv[5]=b.y; rv[6]=b.z; rv[7]=b.w;
        ri[0]=0; ri[1]=1; ri[2]=2; ri[3]=3; ri[4]=4; ri[5]=5; ri[6]=6; ri[7]=7;
        sort8(rv, ri);
    }
    #pragma unroll
    for (int g = 1; g < 8; ++g) {
        float sv[8]; int si[8];
        const float4 a = *(const float4*)(row + g * 8);
        const float4 b = *(const float4*)(row + g * 8 + 4);
        sv[0]=a.x; sv[1]=a.y; sv[2]=a.z; sv[3]=a.w;
        sv[4]=b.x; sv[5]=b.y; sv[6]=b.z; sv[7]=b.w;
        #pragma unroll
        for (int j = 0; j < 8; ++j) si[j] = g * 8 + j;
        sort8(sv, si);
        // Elementwise max of (desc rv, reversed desc sv) = bitonic sequence of the top-8.
        #pragma unroll
        for (int i = 0; i < 8; ++i) {
            const bool c = sv[7 - i] > rv[i];
            rv[i] = c ? sv[7 - i] : rv[i];
            ri[i] = c ? si[7 - i] : ri[i];
        }
        bitonic8(rv, ri);   // rv/ri now sorted descending top-8 so far
    }

    // ---- Softmax over the 8 selected logits (rv[0] is the max) ----
    const float m = rv[0];
    float e[8];
    float sum = 0.f;
    #pragma unroll
    for (int j = 0; j < 8; ++j) { e[j] = __expf(rv[j] - m); sum += e[j]; }
    const float inv = 1.0f / sum;

    // ---- Vectorized stores: idx as int64 (4x b128), weights as f32 (2x b128) ----
    long long* oi = out_idx + (size_t)token * TOP_K;   // 64B aligned
    #pragma unroll
    for (int j = 0; j < 8; j += 2) {
        longlong2 p; p.x = (long long)ri[j]; p.y = (long long)ri[j + 1];
        *(longlong2*)(oi + j) = p;
    }
    float* ow = out_w + (size_t)token * TOP_K;         // 32B aligned
    float4 w0; w0.x = e[0]*inv; w0.y = e[1]*inv; w0.z = e[2]*inv; w0.w = e[3]*inv;
    float4 w1; w1.x = e[4]*inv; w1.y = e[5]*inv; w1.z = e[6]*inv; w1.w = e[7]*inv;
    *(float4*)(ow + 0) = w0;
    *(float4*)(ow + 4) = w1;
}

extern "C" void kernel_launch(void* const* d_in, const int* in_sizes, int n_in,
                              void* d_out, int out_size, void* d_ws, size_t ws_size,
                              hipStream_t stream) {
    (void)n_in; (void)out_size; (void)d_ws; (void)ws_size;

    const float* logits = (const float*)d_in[0];
    const int n_tokens  = in_sizes[0] / N_EXPERTS;   // top_k fixed at 8 per reference

    // d_out layout: [n_tokens*8 x int64 indices] then [n_tokens*8 x f32 weights]
    long long* out_idx = (long long*)d_out;
    float*     out_w   = (float*)((char*)d_out +
                                  (size_t)n_tokens * TOP_K * sizeof(long long));

    const int    blocks = (n_tokens + TOKENS_PER_BLOCK - 1) / TOKENS_PER_BLOCK;
    const size_t shmem  = (size_t)TOKENS_PER_BLOCK * ROW_STRIDE * sizeof(float); // 34816 B

    hipLaunchKernelGGL(topk_router_kernel, dim3(blocks), dim3(BLOCK_THREADS),
                       shmem, stream, logits, out_idx, out_w, n_tokens);
}